// MSDeformableAttention3D_82884278879188
// MI455X (gfx1250) — compile-verified
//
#include <hip/hip_runtime.h>
#include <hip/hip_bf16.h>

// ---------------------------------------------------------------------------
// MS-Deformable-Attention 3D for MI455X (gfx1250, wave32, WMMA)
//
// Stage 1: vproj = value @ W_val + b_val   -> layout (b, h, n, dh)  [WMMA f32]
// Stage 2: off   = (query+query_pos) @ W_off  + b_off               [WMMA f32]
// Stage 3: logit = (query+query_pos) @ W_attn + b_attn              [WMMA f32]
// Stage 4: fused softmax + bilinear sampling + head aggregation
//          (one wave per (b,q,h); lane == dh channel; lane == (l,p) for
//           the 32-wide softmax/offset decode, broadcast via __shfl)
//
// GEMM: each wave owns a 16x64 output strip -> one A fragment feeds 4
// independent v_wmma_f32_16x16x4_f32 per k-step (4x A reuse, latency overlap).
// HAS_A2 / MODE are template params so the hot loop has no runtime branches
// (avoids EXEC save/restore predication around the A2 add).
// ---------------------------------------------------------------------------

#define EMBED 256
#define NH    8
#define NL    4
#define NP    8
#define NZ    4
#define DH    32
#define BS    2
#define NQ    10000
#define NV    14960

typedef __attribute__((ext_vector_type(2))) float v2f;
typedef __attribute__((ext_vector_type(8))) float v8f;

__device__ __constant__ int kH[NL] = {64, 32, 16, 8};
__device__ __constant__ int kW[NL] = {176, 88, 44, 22};
__device__ __constant__ int kS[NL] = {0, 11264, 14080, 14784};

// ---------------------------------------------------------------------------
// GEMM: out = (A [+ A2]) @ W + bias ;  A: MxK row-major, W: KxN row-major.
// One wave computes one 16x64 strip (4 x 16x16 tiles) of C.
// MODE 0: linear out[M][N]
// MODE 1: value-projection permute: row m = b*NV+n, col c = h*32+d
//         -> out[((b*NH+h)*NV + n)*DH + d]
// Requires: M % 16 == 0, N % 64 == 0, K % 4 == 0, NV % 16 == 0 (all hold).
// ---------------------------------------------------------------------------
template <bool HAS_A2, int MODE>
__global__ __launch_bounds__(128)
void msda_gemm_bias(const float* __restrict__ A, const float* __restrict__ A2,
                    const float* __restrict__ W, const float* __restrict__ bias,
                    float* __restrict__ out, int M, int N, int K)
{
    const int lane    = threadIdx.x & 31;
    const int wave    = threadIdx.x >> 5;
    const int stripsN = N >> 6;                       // 16x64 strips along N
    const int stripId = blockIdx.x * 4 + wave;
    if (stripId >= (M >> 4) * stripsN) return;        // wave-uniform guard

    const int m0  = (stripId / stripsN) << 4;
    const int n0  = (stripId % stripsN) << 6;
    const int lh  = lane >> 4;                        // 0: lanes 0-15, 1: 16-31
    const int lm  = lane & 15;
    const int klo = lh * 2;                           // A/B K-pair base per half

    const float* __restrict__ Ar  = A + (size_t)(m0 + lm) * K + klo;
    const float* __restrict__ Ar2 = HAS_A2 ? (A2 + (size_t)(m0 + lm) * K + klo) : nullptr;
    const float* __restrict__ Wc  = W + (size_t)klo * N + (n0 + lm);

    v8f acc0 = {}, acc1 = {}, acc2 = {}, acc3 = {};

#pragma unroll 4
    for (int k = 0; k < K; k += 4) {
        // prefetch the A stream one window ahead (global_prefetch_b8)
        if (k + 64 < K) __builtin_prefetch(Ar + k + 64, 0, 1);

        v2f a;
        if (HAS_A2) {
            a.x = Ar[k]     + Ar2[k];
            a.y = Ar[k + 1] + Ar2[k + 1];
        } else {
            a.x = Ar[k];
            a.y = Ar[k + 1];
        }
        const float* __restrict__ w0 = Wc + (size_t)k * N;       // row k+klo
        const float* __restrict__ w1 = Wc + (size_t)(k + 1) * N; // row k+klo+1
        v2f b0, b1, b2, b3;
        b0.x = w0[0];  b0.y = w1[0];
        b1.x = w0[16]; b1.y = w1[16];
        b2.x = w0[32]; b2.y = w1[32];
        b3.x = w0[48]; b3.y = w1[48];

        acc0 = __builtin_amdgcn_wmma_f32_16x16x4_f32(false, a, false, b0,
                                                     (short)0, acc0, false, false);
        acc1 = __builtin_amdgcn_wmma_f32_16x16x4_f32(false, a, false, b1,
                                                     (short)0, acc1, false, false);
        acc2 = __builtin_amdgcn_wmma_f32_16x16x4_f32(false, a, false, b2,
                                                     (short)0, acc2, false, false);
        acc3 = __builtin_amdgcn_wmma_f32_16x16x4_f32(false, a, false, b3,
                                                     (short)0, acc3, false, false);
    }

    const int rowBase = m0 + lh * 8;   // C/D layout: VGPR r -> M = r (+8 upper lanes)

#pragma unroll
    for (int t = 0; t < 4; ++t) {
        const v8f acc = (t == 0) ? acc0 : (t == 1) ? acc1 : (t == 2) ? acc2 : acc3;
        const int col = n0 + t * 16 + lm;
        const float bv = bias[col];
        if (MODE == 0) {
#pragma unroll
            for (int r = 0; r < 8; ++r)
                out[(size_t)(rowBase + r) * N + col] = acc[r] + bv;
        } else {
            const int h = col >> 5;            // col = h*32 + d
            const int d = col & 31;
#pragma unroll
            for (int r = 0; r < 8; ++r) {
                const int m = rowBase + r;
                const int b = m / NV;          // tiles never straddle batches
                const int n = m - b * NV;
                out[(((size_t)b * NH + h) * NV + n) * DH + d] = acc[r] + bv;
            }
        }
    }
}

// ---------------------------------------------------------------------------
// Fused softmax + bilinear sampling + aggregation.
// One wave per (b,q,h). lane = dh channel for gathers/output.
// lane also = j = l*NP+p (NL*NP == 32 == wave32) for softmax/offset decode.
// ---------------------------------------------------------------------------
__global__ __launch_bounds__(256)
void msda_sampler(const float* __restrict__ vproj,   // (b,h,NV,DH)
                  const float* __restrict__ off,     // (b,q, NH*NL*NP*2)
                  const float* __restrict__ logit,   // (b,q, NH*NL*NP)
                  const float* __restrict__ refp,    // (b,q,NZ,2)
                  float* __restrict__ out)           // (b,q,EMBED)
{
    const int  lane = threadIdx.x & 31;
    const int  wave = threadIdx.x >> 5;
    const long gw   = (long)blockIdx.x * 8 + wave;   // (b*NQ+q)*NH + h
    const int  h    = (int)(gw % NH);
    const long bq   = gw / NH;                       // b*NQ + q
    const int  b    = (int)(bq / NQ);

    // ---- softmax over the 32 (level,point) logits; lane == j ----
    float lg = logit[bq * (NH * NL * NP) + h * (NL * NP) + lane];
    float mx = lg;
#pragma unroll
    for (int s = 16; s; s >>= 1) mx = fmaxf(mx, __shfl_xor(mx, s, 32));
    float e  = __expf(lg - mx);
    float sm = e;
#pragma unroll
    for (int s = 16; s; s >>= 1) sm += __shfl_xor(sm, s, 32);
    const float aw = e / sm;

    // ---- per-(l,p) sampling location; lane == j ----
    const int   lj = lane >> 3;                      // level
    const int   pj = lane & 7;                       // point
    const int   zj = pj & (NZ - 1);                  // Z anchor = p % NZ
    const float ox = off[bq * (NH * NL * NP * 2) + h * (NL * NP * 2) + lane * 2 + 0];
    const float oy = off[bq * (NH * NL * NP * 2) + h * (NL * NP * 2) + lane * 2 + 1];
    const float rx = refp[(bq * NZ + zj) * 2 + 0];
    const float ry = refp[(bq * NZ + zj) * 2 + 1];
    // (rx + ox/w)*w - 0.5 == rx*w + ox - 0.5
    const float px = fmaf(rx, (float)kW[lj], ox) - 0.5f;
    const float py = fmaf(ry, (float)kH[lj], oy) - 0.5f;

    // ---- gather loop: lane == dh channel ----
    const float* __restrict__ vb = vproj + ((size_t)b * NH + h) * ((size_t)NV * DH);
    float accv = 0.0f;

#pragma unroll 4
    for (int j = 0; j < NL * NP; ++j) {
        const float pxj = __shfl(px, j, 32);
        const float pyj = __shfl(py, j, 32);
        const float awj = __shfl(aw, j, 32);
        const int   l   = j >> 3;
        const int   wl  = kW[l];
        const int   hl  = kH[l];
        const float* __restrict__ vlb = vb + (size_t)kS[l] * DH;

        const float x0f = floorf(pxj), y0f = floorf(pyj);
        const float lx  = pxj - x0f,  ly  = pyj - y0f;
        const int   x0  = (int)x0f,   y0  = (int)y0f;

        float s4 = 0.0f;
#pragma unroll
        for (int c = 0; c < 4; ++c) {
            const int   dx = c & 1, dy = c >> 1;
            const int   xi = x0 + dx, yi = y0 + dy;
            const float wx = dx ? lx : (1.0f - lx);
            const float wy = dy ? ly : (1.0f - ly);
            const bool  ok = (xi >= 0) & (xi < wl) & (yi >= 0) & (yi < hl);
            const int   xc = min(max(xi, 0), wl - 1);
            const int   yc = min(max(yi, 0), hl - 1);
            const float v  = vlb[((size_t)yc * wl + xc) * DH + lane]; // 128B/wave
            s4 = fmaf(ok ? wx * wy : 0.0f, v, s4);
        }
        accv = fmaf(awj, s4, accv);
    }

    out[bq * EMBED + h * DH + lane] = accv;
}

// ---------------------------------------------------------------------------
extern "C" void kernel_launch(void* const* d_in, const int* in_sizes, int n_in,
                              void* d_out, int out_size, void* d_ws, size_t ws_size,
                              hipStream_t stream)
{
    (void)in_sizes; (void)n_in; (void)out_size; (void)ws_size;

    const float* query     = (const float*)d_in[0];
    const float* value     = (const float*)d_in[1];
    const float* query_pos = (const float*)d_in[2];
    const float* refp      = (const float*)d_in[3];
    // d_in[4] spatial_shapes / d_in[5] level_start_index: compile-time constants
    const float* W_off     = (const float*)d_in[6];
    const float* b_off     = (const float*)d_in[7];
    const float* W_attn    = (const float*)d_in[8];
    const float* b_attn    = (const float*)d_in[9];
    const float* W_val     = (const float*)d_in[10];
    const float* b_val     = (const float*)d_in[11];
    float* out = (float*)d_out;

    char*  ws     = (char*)d_ws;
    float* vproj  = (float*)ws;                             // BS*NH*NV*DH = 30,638,080 B
    float* offw   = (float*)(ws + (size_t)BS * NH * NV * DH * 4);
    float* logitw = (float*)(ws + (size_t)BS * NH * NV * DH * 4
                                + (size_t)BS * NQ * NH * NL * NP * 2 * 4);

    // Stage 1: value projection -> (b,h,n,dh).  M=29920,N=256,K=256
    {
        const int M = BS * NV, N = EMBED;
        const int strips = (M / 16) * (N / 64);
        msda_gemm_bias<false, 1><<<(strips + 3) / 4, 128, 0, stream>>>(
            value, nullptr, W_val, b_val, vproj, M, N, EMBED);
    }
    // Stage 2: sampling offsets.  M=20000,N=512,K=256
    {
        const int M = BS * NQ, N = NH * NL * NP * 2;
        const int strips = (M / 16) * (N / 64);
        msda_gemm_bias<true, 0><<<(strips + 3) / 4, 128, 0, stream>>>(
            query, query_pos, W_off, b_off, offw, M, N, EMBED);
    }
    // Stage 3: attention logits.  M=20000,N=256,K=256
    {
        const int M = BS * NQ, N = NH * NL * NP;
        const int strips = (M / 16) * (N / 64);
        msda_gemm_bias<true, 0><<<(strips + 3) / 4, 128, 0, stream>>>(
            query, query_pos, W_attn, b_attn, logitw, M, N, EMBED);
    }
    // Stage 4: fused softmax + bilinear sample + aggregate
    {
        const int waves = BS * NQ * NH;   // 160000, 8 waves/block
        msda_sampler<<<waves / 8, 256, 0, stream>>>(vproj, offw, logitw, refp, out);
    }
}